// CorefGRU_58394375356912
// MI455X (gfx1250) — compile-verified
//
#include <hip/hip_runtime.h>

#define B_  256
#define L_  256
#define IN_ 256
#define D_  128
#define H_  64

typedef __attribute__((ext_vector_type(2)))  float        v2f;
typedef __attribute__((ext_vector_type(8)))  float        v8f;
typedef __attribute__((ext_vector_type(4)))  unsigned int v4u;
typedef __attribute__((ext_vector_type(4)))  int          v4i;
typedef __attribute__((ext_vector_type(8)))  int          v8i;

// ---------------------------------------------------------------------------
// TDM 2D tile load: global (row-major, f32) -> LDS.  Descriptor per
// CDNA5 ISA (D# group 0 + group 1, 2D tensor).
// ---------------------------------------------------------------------------
__device__ __forceinline__ void tdm_load_2d(unsigned lds_off_bytes,
                                            const void* gptr,
                                            unsigned elems_per_row,  // tile_dim0 / tensor_dim0
                                            unsigned rows,           // tile_dim1 / tensor_dim1
                                            unsigned row_stride)     // tensor_dim0_stride (elems)
{
    unsigned long long ga = (unsigned long long)(size_t)gptr;
    v4u g0;
    g0[0] = 1u;                                   // count=1, is_restore=0, gather off
    g0[1] = lds_off_bytes;                        // lds_addr
    g0[2] = (unsigned)(ga & 0xFFFFFFFFu);         // global_addr[31:0]
    g0[3] = (unsigned)((ga >> 32) & 0x1FFFFFFu)   // global_addr[56:32]
          | (2u << 30);                           // type = 2 ("image")
    v8i g1;
    g1[0] = (int)(2u << 16);                      // workgroup_mask=0, data_size=2 (4B)
    g1[1] = (int)((elems_per_row & 0xFFFFu) << 16);          // tensor_dim0[15:0]
    g1[2] = (int)(((elems_per_row >> 16) & 0xFFFFu)          // tensor_dim0[31:16]
          |       ((rows & 0xFFFFu) << 16));                 // tensor_dim1[15:0]
    g1[3] = (int)(((rows >> 16) & 0xFFFFu)                   // tensor_dim1[31:16]
          |       ((elems_per_row & 0xFFFFu) << 16));        // tile_dim0
    g1[4] = (int)(rows & 0xFFFFu);                           // tile_dim1 (tile_dim2=0)
    g1[5] = (int)row_stride;                                 // tensor_dim0_stride[31:0]
    g1[6] = 0;                                               // stride hi / dim1_stride lo
    g1[7] = 0;
    v4i z4 = {0, 0, 0, 0};
#if __has_include(<hip/amd_detail/amd_gfx1250_TDM.h>)
    v8i z8 = {0, 0, 0, 0, 0, 0, 0, 0};
    __builtin_amdgcn_tensor_load_to_lds(g0, g1, z4, z4, z8, 0);
#else
    __builtin_amdgcn_tensor_load_to_lds(g0, g1, z4, z4, 0);
#endif
}

__device__ __forceinline__ v8f wmma_f32(v2f a, v2f b, v8f c) {
    return __builtin_amdgcn_wmma_f32_16x16x4_f32(false, a, false, b,
                                                 (short)0, c, false, false);
}

__device__ __forceinline__ float sigmoidf_(float v) {
    return 1.0f / (1.0f + __expf(-v));
}

// ---------------------------------------------------------------------------
// Phase 1: Xg = x @ W_gᵀ + b_g  for g in {r,z,h};  also a = sigmoid(x·(k1-k2)).
// Block = 256 threads (8 waves), computes a 128-row × 128-col slab of one gate.
// LDS: Xs[128][256] (128KB) + Ws[128][256] (128KB), both TDM-loaded.
// Wave w owns M-tile w (16 rows) × all 8 N-tiles; K-loop of V_WMMA_F32_16X16X4.
// ---------------------------------------------------------------------------
__global__ void coref_proj_gemm(const float* __restrict__ x,
                                const float* __restrict__ w_r, const float* __restrict__ b_r,
                                const float* __restrict__ w_z, const float* __restrict__ b_z,
                                const float* __restrict__ w_h, const float* __restrict__ b_h,
                                const float* __restrict__ k1k2,
                                float* __restrict__ Xg_ws, float* __restrict__ a_ws)
{
    extern __shared__ float smem[];
    float* Xs = smem;                 // 128 x 256
    float* Ws = smem + 128 * IN_;     // 128 x 256

    const int gate = blockIdx.y;
    const float* W    = (gate == 0) ? w_r : (gate == 1) ? w_z : w_h;
    const float* bias = (gate == 0) ? b_r : (gate == 1) ? b_z : b_h;
    const int mbase = blockIdx.x * 128;
    const int tid  = threadIdx.x;
    const int lane = tid & 31;
    const int wv   = tid >> 5;

    if (wv == 0) {
        tdm_load_2d(0u, x + (size_t)mbase * IN_, IN_, 128, IN_);
        tdm_load_2d((unsigned)(128 * IN_ * 4), W, IN_, 128, IN_);
        __builtin_amdgcn_s_wait_tensorcnt(0);
    }
    __syncthreads();

    // 2-way softmax gate (only once, on gate==0 blocks), reusing the LDS slab.
    if (gate == 0) {
        const int r = tid >> 1, half = tid & 1;
        float s = 0.0f;
        const float* k1 = k1k2;
        const float* k2 = k1k2 + IN_;
        for (int i = half * 128; i < half * 128 + 128; ++i)
            s += Xs[r * IN_ + i] * (k1[i] - k2[i]);
        s += __shfl_xor(s, 1);
        if (half == 0) a_ws[mbase + r] = sigmoidf_(s);
    }

    const int mrow0 = wv * 16;
    const int lhalf = lane >> 4;      // selects K pair {0,1} vs {2,3}
    const int lmod  = lane & 15;      // row / col within tile

    v8f acc[8] = {};
    for (int k = 0; k < IN_; k += 4) {
        const int kk = k + 2 * lhalf;
        v2f a = *(const v2f*)(Xs + (mrow0 + lmod) * IN_ + kk);
#pragma unroll
        for (int nt = 0; nt < 8; ++nt) {
            v2f b = *(const v2f*)(Ws + (nt * 16 + lmod) * IN_ + kk);
            acc[nt] = wmma_f32(a, b, acc[nt]);
        }
    }

    float* Y = Xg_ws + (size_t)gate * ((size_t)B_ * L_ * D_);
#pragma unroll
    for (int nt = 0; nt < 8; ++nt) {
        const int n  = nt * 16 + lmod;
        const float bb = bias[n];
#pragma unroll
        for (int i = 0; i < 8; ++i) {
            const int m = mbase + mrow0 + i + 8 * lhalf;
            Y[(size_t)m * D_ + n] = acc[nt][i] + bb;
        }
    }
}

// ---------------------------------------------------------------------------
// Phase 2: sequential coref-GRU scan.  Block = 256 threads (8 waves) owns a
// tile of 16 independent batch chains for all L steps.  u_r/u_z/u_h are
// TDM-loaded into LDS once; each wave then hoists its invariant 16 columns
// of all three U matrices into registers (32 k-steps x v2f x 3 = 192 VGPRs).
// __launch_bounds__(256, 1): this kernel is 1 workgroup/WGP by LDS anyway
// (208 KB of 320 KB), so let the wave take ~300 VGPRs instead of spilling —
// CDNA5 allows up to 1024 VGPRs/wave.  The serial t-loop then touches LDS
// only for the fresh A fragments of m_t (32 ds_load_b64 + 96 WMMA per step),
// with the step's Xr/Xz/Xh global loads issued before the WMMA chain so
// their latency overlaps matrix work.  History buf[i] == out[b, i-1, 64:],
// read back from d_out with a fence (same workgroup wrote it).
// ---------------------------------------------------------------------------
__global__ void __launch_bounds__(256, 1)
coref_gru_scan(const float* __restrict__ Xg_ws,
               const float* __restrict__ a_ws,
               const int*   __restrict__ cor,
               const float* __restrict__ u_r,
               const float* __restrict__ u_z,
               const float* __restrict__ u_h,
               float* __restrict__ out)
{
    extern __shared__ float smem[];
    float* Ur = smem;                        // 128x128
    float* Uz = smem + D_ * D_;              // 128x128
    float* Uh = smem + 2 * D_ * D_;          // 128x128
    float* mt = smem + 3 * D_ * D_;          // 16x128
    float* hp = smem + 3 * D_ * D_ + 16*D_;  // 16x128 (h_prev)

    const int tid   = threadIdx.x;
    const int lane  = tid & 31;
    const int wv    = tid >> 5;
    const int bbase = blockIdx.x * 16;

    if (wv == 0) {
        tdm_load_2d(0u,                          u_r, D_, D_, D_);
        tdm_load_2d((unsigned)(D_ * D_ * 4),     u_z, D_, D_, D_);
        tdm_load_2d((unsigned)(2 * D_ * D_ * 4), u_h, D_, D_, D_);
        __builtin_amdgcn_s_wait_tensorcnt(0);
    }
    for (int e = tid; e < 16 * D_; e += blockDim.x) hp[e] = 0.0f;
    __syncthreads();

    const float* Xr = Xg_ws;
    const float* Xz = Xg_ws + (size_t)B_ * L_ * D_;
    const float* Xh = Xg_ws + 2 * (size_t)B_ * L_ * D_;

    const int lhalf = lane >> 4;
    const int lmod  = lane & 15;
    const int n0    = wv * 16;

    // ---- hoist this wave's B-fragments of u_r/u_z/u_h into registers ----
    v2f bfr[32], bfz[32], bfh[32];
#pragma unroll
    for (int ks = 0; ks < 32; ++ks) {
        const int kk = 4 * ks + 2 * lhalf;
        bfr[ks] = *(const v2f*)(Ur + (n0 + lmod) * D_ + kk);
        bfz[ks] = *(const v2f*)(Uz + (n0 + lmod) * D_ + kk);
        bfh[ks] = *(const v2f*)(Uh + (n0 + lmod) * D_ + kk);
    }

    for (int t = 0; t < L_; ++t) {
        // ---- build m_t = [a*h_prev[:, :H] | (1-a)*buf[i_eff]] into LDS ----
        for (int e = tid; e < 16 * D_; e += blockDim.x) {
            const int b = e >> 7;
            const int j = e & (D_ - 1);
            const float at = a_ws[(size_t)(bbase + b) * L_ + t];
            float v;
            if (j < H_) {
                v = at * hp[b * D_ + j];
            } else {
                const int ct = cor[(size_t)(bbase + b) * L_ + t];
                const int ie = (ct == 0) ? t : ct;
                float m2 = 0.0f;
                if (ie > 0)
                    m2 = out[((size_t)(bbase + b) * L_ + (ie - 1)) * D_ + j];
                v = (1.0f - at) * m2;
            }
            mt[e] = v;
        }
        __syncthreads();

        // ---- prefetch this step's input projections (overlap with WMMA) ----
        const int n = n0 + lmod;
        float xrv[8], xzv[8], xhv[8];
#pragma unroll
        for (int i = 0; i < 8; ++i) {
            const size_t row = (size_t)(bbase + i + 8 * lhalf) * L_ + t;
            xrv[i] = Xr[row * D_ + n];
            xzv[i] = Xz[row * D_ + n];
            xhv[i] = Xh[row * D_ + n];
        }

        // ---- three 16x128 @ 128x16 WMMA matmuls (B operands in registers) --
        v8f aR = {}, aZ = {}, aH = {};
#pragma unroll
        for (int ks = 0; ks < 32; ++ks) {
            v2f a = *(const v2f*)(mt + lmod * D_ + 4 * ks + 2 * lhalf);
            aR = wmma_f32(a, bfr[ks], aR);
            aZ = wmma_f32(a, bfz[ks], aZ);
            aH = wmma_f32(a, bfh[ks], aH);
        }

        // ---- gate nonlinearity + state update ----
#pragma unroll
        for (int i = 0; i < 8; ++i) {
            const int m = i + 8 * lhalf;        // batch within tile
            const size_t row = (size_t)(bbase + m) * L_ + t;
            const float mm = mt[m * D_ + n];
            const float r  = sigmoidf_(xrv[i] + aR[i]);
            const float z  = sigmoidf_(xzv[i] + aZ[i]);
            const float hh = tanhf(xhv[i] + r * aH[i]);
            const float h  = (1.0f - z) * mm + z * hh;
            out[row * D_ + n] = h;
            hp[m * D_ + n]    = h;
        }
        __threadfence();   // make out[.,t,64:] visible for next step's coref read
        __syncthreads();
    }
}

// ---------------------------------------------------------------------------
extern "C" void kernel_launch(void* const* d_in, const int* in_sizes, int n_in,
                              void* d_out, int out_size, void* d_ws, size_t ws_size,
                              hipStream_t stream)
{
    const float* x    = (const float*)d_in[0];
    const int*   cor  = (const int*)  d_in[1];
    const float* w_r  = (const float*)d_in[2];
    const float* b_r  = (const float*)d_in[3];
    const float* u_r  = (const float*)d_in[4];
    const float* w_z  = (const float*)d_in[5];
    const float* b_z  = (const float*)d_in[6];
    const float* u_z  = (const float*)d_in[7];
    const float* w_h  = (const float*)d_in[8];
    const float* b_h  = (const float*)d_in[9];
    const float* u_h  = (const float*)d_in[10];
    const float* k1k2 = (const float*)d_in[11];
    float* out = (float*)d_out;

    float* Xg   = (float*)d_ws;                       // 3 * B*L*D f32
    float* a_ws = Xg + 3 * (size_t)B_ * L_ * D_;      // B*L f32

    const size_t lds1 = (size_t)2 * 128 * IN_ * 4;                 // 256 KB
    coref_proj_gemm<<<dim3((B_ * L_) / 128, 3), 256, lds1, stream>>>(
        x, w_r, b_r, w_z, b_z, w_h, b_h, k1k2, Xg, a_ws);

    const size_t lds2 = (size_t)(3 * D_ * D_ + 2 * 16 * D_) * 4;   // ~208 KB
    coref_gru_scan<<<dim3(B_ / 16), 256, lds2, stream>>>(
        Xg, a_ws, cor, u_r, u_z, u_h, out);
}